// RGCN_10806137716921
// MI455X (gfx1250) — compile-verified
//
#include <hip/hip_runtime.h>

// RGCN 2-layer, MI455X (gfx1250, wave32).
// Strategy: per-relation transform-then-scatter so the per-relation
// intermediate t_r ([N,128] = 25.6MB) is L2-resident (192MB L2) -> all edge
// gathers and atomic scatter-adds are L2 traffic, not HBM. fp32 WMMA
// (v_wmma_f32_16x16x4_f32) keeps full reference precision on the tensor path.
// Workspace: N*128 (hsum) + N*128 (t_r) floats = ~51.2 MB.

typedef __attribute__((ext_vector_type(2))) float v2f;
typedef __attribute__((ext_vector_type(8))) float v8f;

__global__ __launch_bounds__(256)
void fill_zero_kernel(float* __restrict__ p, int n) {
    int i = blockIdx.x * blockDim.x + threadIdx.x;
    if (i < n) p[i] = 0.0f;
}

// T[nrows, NCOL] = (RELU ? relu(A) : A)[nrows, 128] @ W[128, NCOL]
// Each wave computes a 16-row strip across all NCOL/16 column tiles.
template <int NCOL, bool RELU>
__global__ __launch_bounds__(256)
void rgcn_gemm_kernel(const float* __restrict__ A, const float* __restrict__ W,
                      float* __restrict__ T, int nrows) {
    constexpr int K  = 128;
    constexpr int NT = NCOL / 16;
    __shared__ float w_lds[K * NCOL];

    // Stage W into LDS. XOR-16 column swizzle on rows with (row & 2) set:
    // the two wave halves read rows k0+{0,1} vs k0+{2,3}; with stride NCOL
    // (128) 2*stride = 0 mod 64 banks -> swizzle makes halves bank-disjoint.
    for (int idx = threadIdx.x; idx < K * NCOL; idx += blockDim.x) {
        int row = idx / NCOL, col = idx % NCOL;
        int scol = (row & 2) ? (col ^ 16) : col;
        w_lds[row * NCOL + scol] = W[idx];
    }
    __syncthreads();

    const int lane = threadIdx.x & 31;
    const int wave = threadIdx.x >> 5;
    const int m0   = (blockIdx.x * 8 + wave) * 16;
    if (m0 >= nrows) return;

    const int mc   = lane & 15;   // A row-in-tile / B,C column-in-tile
    const int half = lane >> 4;   // 0 or 1
    const int kh   = half * 2;    // A/B K sub-pair per ISA 16x4 layout
    const int cxor = half << 4;   // undo LDS swizzle for upper half

    const v8f vzero = {};
    v8f acc[NT];
    #pragma unroll
    for (int t = 0; t < NT; ++t) acc[t] = vzero;

    const float* arow = A + (size_t)(m0 + mc) * K;
    for (int k0 = 0; k0 < K; k0 += 4) {
        // A fragment: lane holds A[m0+mc][k0+kh .. k0+kh+1] (contiguous 8B)
        v2f a = *(const v2f*)(arow + k0 + kh);
        if (RELU) { a.x = fmaxf(a.x, 0.0f); a.y = fmaxf(a.y, 0.0f); }
        #pragma unroll
        for (int t = 0; t < NT; ++t) {
            v2f b;
            b.x = w_lds[(k0 + kh)     * NCOL + ((t * 16 + mc) ^ cxor)];
            b.y = w_lds[(k0 + kh + 1) * NCOL + ((t * 16 + mc) ^ cxor)];
            acc[t] = __builtin_amdgcn_wmma_f32_16x16x4_f32(
                false, a, false, b, (short)0, acc[t], false, false);
        }
    }

    // C/D layout: VGPR v, lanes0-15 -> M=v, lanes16-31 -> M=v+8, N=mc
    const int srow = m0 + half * 8;
    #pragma unroll
    for (int t = 0; t < NT; ++t) {
        #pragma unroll
        for (int v = 0; v < 8; ++v) {
            T[(size_t)(srow + v) * NCOL + t * 16 + mc] = acc[t][v];
        }
    }
}

// One wave per edge. Wave-uniform etype filter; coalesced row gather from the
// L2-resident t_r, scaled by norm, atomic-added into out[dst].
template <int DIM>
__global__ __launch_bounds__(256)
void rgcn_scatter_kernel(const float* __restrict__ T, const int* __restrict__ src,
                         const int* __restrict__ dst, const int* __restrict__ etypes,
                         const float* __restrict__ norm, float* __restrict__ out,
                         int nedges, int rel) {
    int e = blockIdx.x * (blockDim.x >> 5) + (threadIdx.x >> 5);
    if (e >= nedges) return;
    if (etypes[e] != rel) return;   // uniform across the wave
    const int lane = threadIdx.x & 31;
    const float nv = norm[e];
    const float* trow = T + (size_t)src[e] * DIM;
    float* orow = out + (size_t)dst[e] * DIM;
    #pragma unroll
    for (int i = 0; i < DIM / 32; ++i) {
        atomicAdd(&orow[lane + 32 * i], trow[lane + 32 * i] * nv);
    }
}

extern "C" void kernel_launch(void* const* d_in, const int* in_sizes, int n_in,
                              void* d_out, int out_size, void* d_ws, size_t ws_size,
                              hipStream_t stream) {
    const float* feat = (const float*)d_in[0];
    const float* norm = (const float*)d_in[1];
    const float* W1   = (const float*)d_in[2];
    const float* W2   = (const float*)d_in[3];
    const int*   src  = (const int*)d_in[4];
    const int*   dst  = (const int*)d_in[5];
    const int*   et   = (const int*)d_in[6];
    float* out = (float*)d_out;

    const int IN = 128, H = 128, O = 64;
    const int N = in_sizes[0] / IN;          // 50000
    const int E = in_sizes[1];               // 800000
    const int R = in_sizes[2] / (IN * H);    // 16

    float* hsum = (float*)d_ws;              // [N, H] layer-1 accumulator
    float* t_r  = hsum + (size_t)N * H;      // [N, max(H,O)] per-relation scratch

    const int gemm_grid = ((N + 15) / 16 + 7) / 8;
    const int scat_grid = (E + 7) / 8;

    // Layer 1: hsum = sum_r scatter(feat @ W1[r])
    fill_zero_kernel<<<(N * H + 255) / 256, 256, 0, stream>>>(hsum, N * H);
    for (int r = 0; r < R; ++r) {
        rgcn_gemm_kernel<128, false><<<gemm_grid, 256, 0, stream>>>(
            feat, W1 + (size_t)r * IN * H, t_r, N);
        rgcn_scatter_kernel<128><<<scat_grid, 256, 0, stream>>>(
            t_r, src, dst, et, norm, hsum, E, r);
    }

    // Layer 2: out = sum_r scatter(relu(hsum) @ W2[r])  (relu fused into GEMM A-load)
    fill_zero_kernel<<<(N * O + 255) / 256, 256, 0, stream>>>(out, N * O);
    for (int r = 0; r < R; ++r) {
        rgcn_gemm_kernel<64, true><<<gemm_grid, 256, 0, stream>>>(
            hsum, W2 + (size_t)r * H * O, t_r, N);
        rgcn_scatter_kernel<64><<<scat_grid, 256, 0, stream>>>(
            t_r, src, dst, et, norm, out, E, r);
    }
}